// OutputPPBlockSWM_32384053412129
// MI455X (gfx1250) — compile-verified
//
#include <hip/hip_runtime.h>
#include <math.h>
#include <stdint.h>

#define H       128
#define RDIM    6
#define OE      256
#define NHEADS  2
#define NLAYERS 3
#define ALPHA   0.5f
#define MTILE   32
#define EDGES_PER_BLOCK 8

typedef __attribute__((ext_vector_type(2))) float v2f;
typedef __attribute__((ext_vector_type(8))) float v8f;

// ---------------- zero workspace ----------------
__global__ void zero_f32(float* __restrict__ p, int n) {
  int t = blockIdx.x * blockDim.x + threadIdx.x;
  if (t < n) p[t] = 0.f;
}

// ---------------- edge phase: h = (rbf@W_rbf)*x*w, scatter-add ----------------
__global__ void __launch_bounds__(256) edge_scatter(
    const float* __restrict__ x,
    const float* __restrict__ rbf,
    const float* __restrict__ lambda_d,
    const float* __restrict__ W_rbf,
    const int*   __restrict__ idx,
    const int*   __restrict__ bt,
    float*       __restrict__ xspe,
    int E) {
  const int j   = threadIdx.x & 127;   // channel 0..127
  const int grp = threadIdx.x >> 7;    // 0/1

  float wr[RDIM];
#pragma unroll
  for (int r = 0; r < RDIM; ++r) wr[r] = W_rbf[r * H + j];

  int base = blockIdx.x * EDGES_PER_BLOCK + grp;
#pragma unroll
  for (int it = 0; it < EDGES_PER_BLOCK / 2; ++it) {
    int e = base + it * 2;
    if (e < E) {
      float dot = 0.f;
#pragma unroll
      for (int r = 0; r < RDIM; ++r) dot += rbf[e * RDIM + r] * wr[r];
      float w   = ALPHA + (1.f - ALPHA) * lambda_d[bt[e]];
      float val = dot * x[(long)e * H + j] * w;
      int node  = idx[e];
      unsafeAtomicAdd(&xspe[(long)node * H + j], val);  // native f32 atomic, L2-resident
    }
  }
}

// ---------------- MLP phase ----------------
// grid = (ceil(N/32), NHEADS), block = 256 (8 waves). f32 WMMA 16x16x4.
// Each wave owns 2 column tiles; B fragments are loaded ONCE per k-step and
// fed to two accumulators (row subtiles 0-15 / 16-31).
__global__ void __launch_bounds__(256) mlp_kernel(
    const float* __restrict__ xspe,
    const float* __restrict__ W_up,
    const float* __restrict__ Ws,
    const float* __restrict__ bs,
    const float* __restrict__ W_out,
    float*       __restrict__ out,
    int N) {
  extern __shared__ float smem[];
  float* sIn = smem;                 // [MTILE][132]  input tile
  float* sA  = sIn + MTILE * 132;    // [MTILE][260]  ping
  float* sB  = sA + MTILE * 260;     // [MTILE][260]  pong

  const int a         = blockIdx.y;
  const long nodeBase = (long)blockIdx.x * MTILE;

  const int tid  = threadIdx.x;
  const int lane = tid & 31;
  const int wid  = tid >> 5;        // wave 0..7
  const int half = lane >> 4;       // 0/1
  const int l16  = lane & 15;

  // ---- input tile: 32x128 f32 (16KB) ----
  if (nodeBase + MTILE <= (long)N) {
    // fast path: CDNA5 async copy global->LDS, 4 x b128 per thread
    const float* src = xspe + nodeBase * H;
#pragma unroll
    for (int it = 0; it < 4; ++it) {
      int q  = tid + it * 256;        // b128 chunk id, 32 chunks per row
      int r  = q >> 5;
      int c4 = (q & 31) * 4;
      uint32_t ldsOff = (uint32_t)(uintptr_t)(&sIn[r * 132 + c4]);
      uint32_t gOff   = (uint32_t)((r * H + c4) * 4);
      asm volatile("global_load_async_to_lds_b128 %0, %1, %2"
                   :
                   : "v"(ldsOff), "v"(gOff), "s"(src)
                   : "memory");
    }
    asm volatile("s_wait_asynccnt 0x0" ::: "memory");
  } else {
    // tail block: guarded synchronous loads (zeros past N)
    for (int p = tid; p < MTILE * H; p += 256) {
      int r = p >> 7, c = p & (H - 1);
      long node = nodeBase + r;
      sIn[r * 132 + c] = (node < (long)N) ? xspe[node * H + c] : 0.f;
    }
  }
  __syncthreads();

  const float* Wu = W_up + (long)a * H * OE;

  // ---- up projection: sIn[32x128] @ Wu[128x256] -> sA (no bias/act) ----
  for (int cti = 0; cti < 2; ++cti) {
    const int ct  = wid + cti * 8;
    const int col = ct * 16 + l16;
    v8f acc0 = {}, acc1 = {};
    for (int k0 = 0; k0 < H; k0 += 4) {
      const int k = k0 + 2 * half;
      v2f a0, a1, bv;
      const float* ap0 = &sIn[l16 * 132 + k];
      const float* ap1 = &sIn[(16 + l16) * 132 + k];
      a0.x = ap0[0]; a0.y = ap0[1];
      a1.x = ap1[0]; a1.y = ap1[1];
      bv.x = Wu[(long)k * OE + col];
      bv.y = Wu[(long)(k + 1) * OE + col];
      acc0 = __builtin_amdgcn_wmma_f32_16x16x4_f32(
          false, a0, false, bv, (short)0, acc0, false, false);
      acc1 = __builtin_amdgcn_wmma_f32_16x16x4_f32(
          false, a1, false, bv, (short)0, acc1, false, false);
    }
#pragma unroll
    for (int v = 0; v < 8; ++v) {
      int row = v + 8 * half;                      // D layout: VGPR v -> M=v / v+8
      sA[row * 260 + col]        = acc0[v];
      sA[(16 + row) * 260 + col] = acc1[v];
    }
  }
  __syncthreads();

  // ---- hidden layers: y = silu(y @ Ws[a,l] + bs[a,l]) ----
  float* bufs[2] = { sA, sB };
  for (int l = 0; l < NLAYERS; ++l) {
    const float* Wl = Ws + ((long)a * NLAYERS + l) * OE * OE;
    const float* bl = bs + ((long)a * NLAYERS + l) * OE;
    const float* in = bufs[l & 1];
    float*       ob = bufs[(l + 1) & 1];
    for (int cti = 0; cti < 2; ++cti) {
      const int ct  = wid + cti * 8;
      const int col = ct * 16 + l16;
      v8f acc0 = {}, acc1 = {};
      for (int k0 = 0; k0 < OE; k0 += 4) {
        const int k = k0 + 2 * half;
        v2f a0, a1, bv;
        const float* ap0 = &in[l16 * 260 + k];
        const float* ap1 = &in[(16 + l16) * 260 + k];
        a0.x = ap0[0]; a0.y = ap0[1];
        a1.x = ap1[0]; a1.y = ap1[1];
        bv.x = Wl[(long)k * OE + col];
        bv.y = Wl[(long)(k + 1) * OE + col];
        acc0 = __builtin_amdgcn_wmma_f32_16x16x4_f32(
            false, a0, false, bv, (short)0, acc0, false, false);
        acc1 = __builtin_amdgcn_wmma_f32_16x16x4_f32(
            false, a1, false, bv, (short)0, acc1, false, false);
      }
      const float bias = bl[col];
#pragma unroll
      for (int v = 0; v < 8; ++v) {
        int row = v + 8 * half;
        float y0 = acc0[v] + bias;
        float y1 = acc1[v] + bias;
        y0 = y0 / (1.f + __expf(-y0));             // SiLU
        y1 = y1 / (1.f + __expf(-y1));
        ob[row * 260 + col]        = y0;
        ob[(16 + row) * 260 + col] = y1;
      }
    }
    __syncthreads();
  }

  // after 3 layers the live buffer is sB (sA->sB->sA->sB)
  const float* fin = sB;
  const float* Wo  = W_out + (long)a * OE;

  // ---- out projection (256 -> 1): 8 chunks x 32, tree-reduce via LDS ----
  {
    const int nl = tid & 31;
    const int ch = tid >> 5;
    float p = 0.f;
#pragma unroll
    for (int k = 0; k < 32; ++k) {
      int c = ch * 32 + k;
      p += fin[nl * 260 + c] * Wo[c];
    }
    sIn[nl * 8 + ch] = p;                          // reuse sIn as scratch
  }
  __syncthreads();
  if (tid < MTILE) {
    long node = nodeBase + tid;
    if (node < (long)N) {
      float s = 0.f;
#pragma unroll
      for (int ch = 0; ch < 8; ++ch) s += sIn[tid * 8 + ch];
      out[(long)a * N + node] = s;
    }
  }
}

extern "C" void kernel_launch(void* const* d_in, const int* in_sizes, int n_in,
                              void* d_out, int out_size, void* d_ws, size_t ws_size,
                              hipStream_t stream) {
  const float* x     = (const float*)d_in[0];
  const float* rbf   = (const float*)d_in[1];
  const float* lam   = (const float*)d_in[2];
  const float* W_rbf = (const float*)d_in[3];
  const float* W_up  = (const float*)d_in[4];
  const float* Ws    = (const float*)d_in[5];
  const float* bs    = (const float*)d_in[6];
  const float* W_out = (const float*)d_in[7];
  const int*   idx   = (const int*)d_in[8];
  const int*   bt    = (const int*)d_in[9];
  float* out = (float*)d_out;

  const int E = in_sizes[8];           // edges
  const int N = out_size / NHEADS;     // OC == 1
  float* xspe = (float*)d_ws;          // N*H f32 accumulator (L2-resident)

  const int nz = N * H;
  zero_f32<<<(nz + 255) / 256, 256, 0, stream>>>(xspe, nz);

  edge_scatter<<<(E + EDGES_PER_BLOCK - 1) / EDGES_PER_BLOCK, 256, 0, stream>>>(
      x, rbf, lam, W_rbf, idx, bt, xspe, E);

  dim3 g((N + MTILE - 1) / MTILE, NHEADS);
  size_t shmem = (size_t)(MTILE * 132 + 2 * MTILE * 260) * sizeof(float);
  mlp_kernel<<<g, 256, shmem, stream>>>(xspe, W_up, Ws, bs, W_out, out, N);
}